// Loss_AdaptiveClustering_39230231282168
// MI455X (gfx1250) — compile-verified
//
#include <hip/hip_runtime.h>

typedef __attribute__((ext_vector_type(2))) float v2f;
typedef __attribute__((ext_vector_type(8))) float v8f;

// ---------------------------------------------------------------------------
// Zero the accumulator head of the workspace (cls acc + per-class sums/counts)
// ---------------------------------------------------------------------------
__global__ __launch_bounds__(256) void init_ws(float* __restrict__ ws, int n) {
    int i = blockIdx.x * blockDim.x + threadIdx.x;
    if (i < n) ws[i] = 0.0f;
}

// ---------------------------------------------------------------------------
// loss_cls partial: sum over (o - onehot(y))^2, grid-stride + block reduce
// ---------------------------------------------------------------------------
__global__ __launch_bounds__(256) void cls_kernel(const float* __restrict__ o,
                                                  const int* __restrict__ y,
                                                  float* __restrict__ acc,
                                                  int B, int C) {
    __shared__ float red[256];
    const int tid = threadIdx.x;
    const int idx = blockIdx.x * blockDim.x + tid;
    const int stride = gridDim.x * blockDim.x;
    const int N = B * C;
    float s = 0.0f;
    for (int i = idx; i < N; i += stride) {
        int b = i / C;
        int c = i - b * C;
        float v = o[i] - ((y[b] == c) ? 1.0f : 0.0f);
        s = fmaf(v, v, s);
    }
    red[tid] = s;
    __syncthreads();
    for (int off = 128; off > 0; off >>= 1) {
        if (tid < off) red[tid] += red[tid + off];
        __syncthreads();
    }
    if (tid == 0) atomicAdd(acc, red[0]);
}

// ---------------------------------------------------------------------------
// loss_close partial: one 128-thread block per sample; gather only z[i,y_i,:]
// (4 MB of the 419 MB tensor). Per-class segment sums + counts via atomics.
// ---------------------------------------------------------------------------
__global__ __launch_bounds__(128) void close_kernel(const float* __restrict__ z,
                                                    const float* __restrict__ W,
                                                    const int* __restrict__ y,
                                                    float* __restrict__ sums,
                                                    float* __restrict__ counts,
                                                    int C, int D) {
    __shared__ float red[128];
    const int i = blockIdx.x;
    const int t = threadIdx.x;
    const int c = y[i];
    const float d = z[((size_t)i * C + c) * D + t] - W[(size_t)c * D + t];
    red[t] = d * d;
    __syncthreads();
    for (int off = 64; off > 0; off >>= 1) {
        if (t < off) red[t] += red[t + off];
        __syncthreads();
    }
    if (t == 0) {
        atomicAdd(&sums[c], red[0]);
        atomicAdd(&counts[c], 1.0f);
    }
}

// ---------------------------------------------------------------------------
// Gram matrix G = W * W^T via V_WMMA_F32_16X16X4_F32.
// One wave per 16x16 tile; K = D in steps of 4.
// A-matrix layout (16x4 f32): lane l -> M = l&15; VGPR0 = A[M][k0+2*(l>>4)],
// VGPR1 = A[M][k0+2*(l>>4)+1]. B (=W^T) uses the mirrored placement, so both
// operands share the same lane->K mapping and the dot product is exact for
// any internal K ordering. Rows >= C are zero-masked with cndmask (no EXEC
// divergence: WMMA requires EXEC all-ones).
// C/D layout: lane l -> N = l&15, M = r + 8*(l>>4) for VGPR r.
// ---------------------------------------------------------------------------
__global__ __launch_bounds__(32) void gram_wmma(const float* __restrict__ W,
                                                float* __restrict__ G,
                                                int C, int D, int MT, int PAD) {
    const int tile = blockIdx.x;
    const int mt = tile / MT;
    const int nt = tile - mt * MT;
    const int lane = threadIdx.x;
    const int half = lane >> 4;
    const int l16  = lane & 15;

    const int rowA = mt * 16 + l16;
    const int rowB = nt * 16 + l16;
    const float maskA = (rowA < C) ? 1.0f : 0.0f;
    const float maskB = (rowB < C) ? 1.0f : 0.0f;
    const float* pa = W + (size_t)((rowA < C) ? rowA : 0) * D;
    const float* pb = W + (size_t)((rowB < C) ? rowB : 0) * D;

    v8f acc = {0.f, 0.f, 0.f, 0.f, 0.f, 0.f, 0.f, 0.f};
    for (int k0 = 0; k0 < D; k0 += 4) {
        const int ka = k0 + 2 * half;
        v2f a = *(const v2f*)(pa + ka);
        v2f b = *(const v2f*)(pb + ka);
        a *= maskA;
        b *= maskB;
        acc = __builtin_amdgcn_wmma_f32_16x16x4_f32(
            /*neg_a=*/false, a, /*neg_b=*/false, b,
            /*c_mod=*/(short)0, acc, /*reuse_a=*/false, /*reuse_b=*/false);
    }

    const int ncol  = nt * 16 + l16;
    const int mbase = mt * 16 + half * 8;
#pragma unroll
    for (int r = 0; r < 8; ++r)
        G[(size_t)(mbase + r) * PAD + ncol] = acc[r];
}

// ---------------------------------------------------------------------------
// Assemble the four scalars. Uses diag(G) and row-sums of G:
//   S_c = (C*n_c + sum_j n_j - 2*rowsum_c) / ((C-1)*D)
//   loss_dist = sum_c relu(1 - S_c) / C
// ---------------------------------------------------------------------------
__global__ __launch_bounds__(128) void final_kernel(const float* __restrict__ G,
                                                    const float* __restrict__ sums,
                                                    const float* __restrict__ counts,
                                                    const float* __restrict__ acc_cls,
                                                    float* __restrict__ out,
                                                    int B, int C, int D, int PAD) {
    __shared__ float red[128];
    __shared__ float Tsh, distsh;
    const int t = threadIdx.x;

    float n_c = 0.f, R_c = 0.f, closec = 0.f;
    if (t < C) {
        const float* row = G + (size_t)t * PAD;
        n_c = row[t];
        float s = 0.f;
        for (int j = 0; j < C; ++j) s += row[j];
        R_c = s;
        closec = sums[t] / (counts[t] * (float)D);
    }

    // T = sum_c n_c
    red[t] = n_c; __syncthreads();
    for (int off = 64; off > 0; off >>= 1) { if (t < off) red[t] += red[t + off]; __syncthreads(); }
    if (t == 0) Tsh = red[0];
    __syncthreads();
    const float T = Tsh;

    float relu = 0.f;
    if (t < C) {
        float S = ((float)C * n_c + T - 2.0f * R_c) / ((float)(C - 1) * (float)D);
        relu = fmaxf(1.0f - S, 0.0f);   // MARGIN = 1.0
    }
    red[t] = relu; __syncthreads();
    for (int off = 64; off > 0; off >>= 1) { if (t < off) red[t] += red[t + off]; __syncthreads(); }
    if (t == 0) distsh = red[0] / (float)C;
    __syncthreads();

    red[t] = closec; __syncthreads();
    for (int off = 64; off > 0; off >>= 1) { if (t < off) red[t] += red[t + off]; __syncthreads(); }
    if (t == 0) {
        const float l_cls   = acc_cls[0] / ((float)B * (float)C);
        const float l_close = red[0];
        const float l_dist  = distsh;
        out[0] = l_cls + l_close + l_dist;
        out[1] = l_cls;
        out[2] = l_close;
        out[3] = l_dist;
    }
}

// ---------------------------------------------------------------------------
extern "C" void kernel_launch(void* const* d_in, const int* in_sizes, int n_in,
                              void* d_out, int out_size, void* d_ws, size_t ws_size,
                              hipStream_t stream) {
    const float* z = (const float*)d_in[0];   // [B, C, D]
    const float* W = (const float*)d_in[1];   // [C, D]
    const float* o = (const float*)d_in[2];   // [B, C]
    const int*   y = (const int*)d_in[3];     // [B]
    float* out = (float*)d_out;               // 4 scalars
    float* wsf = (float*)d_ws;

    const int B = in_sizes[3];
    const int C = in_sizes[2] / B;            // 100
    const int D = in_sizes[1] / C;            // 128
    const int MT  = (C + 15) / 16;            // 7
    const int PAD = MT * 16;                  // 112

    // Workspace layout (floats): [0]=cls acc, [1..C]=sums, [1+C..1+2C]=counts,
    // [256..256+PAD*PAD) = padded Gram matrix G. ~51 KB total.
    float* acc_cls = wsf + 0;
    float* sums    = wsf + 1;
    float* counts  = wsf + 1 + C;
    float* G       = wsf + 256;

    init_ws<<<1, 256, 0, stream>>>(wsf, 256);
    cls_kernel<<<512, 256, 0, stream>>>(o, y, acc_cls, B, C);
    close_kernel<<<B, D, 0, stream>>>(z, W, y, sums, counts, C, D);
    gram_wmma<<<MT * MT, 32, 0, stream>>>(W, G, C, D, MT, PAD);
    final_kernel<<<1, 128, 0, stream>>>(G, sums, counts, acc_cls, out, B, C, D, PAD);
}